// RelationalKENN_11038065951415
// MI455X (gfx1250) — compile-verified
//
#include <hip/hip_runtime.h>
#include <stdint.h>

#define N_NODES 50000
#define N_UNARY 16
#define N_EDGES 1600000

#define EBLOCK 256
#define EPT 2
#define TILE_EDGES (EBLOCK * EPT)        // 512 edges per block
#define TILE_ELEMS (TILE_EDGES * 4)      // 2048 f32 = 8 KB per block

typedef __attribute__((ext_vector_type(4))) unsigned int u32x4;
typedef __attribute__((ext_vector_type(8))) int          i32x8;
typedef __attribute__((ext_vector_type(4))) int          i32x4;

__device__ __forceinline__ void softmax3(float l0, float l1, float l2, float w,
                                         float& d0, float& d1, float& d2) {
    float m  = fmaxf(l0, fmaxf(l1, l2));
    float e0 = __expf(l0 - m);
    float e1 = __expf(l1 - m);
    float e2 = __expf(l2 - m);
    float inv = w / (e0 + e1 + e2);
    d0 = e0 * inv; d1 = e1 * inv; d2 = e2 * inv;
}

// Stage 1: u = unary + KE(unary); write to out_u (accumulator base) and ws_u (clean gather copy)
__global__ void kenn_node_kernel(const float* __restrict__ unary,
                                 const float* __restrict__ wu,
                                 float* __restrict__ out_u,
                                 float* __restrict__ ws_u) {
    int row = blockIdx.x * blockDim.x + threadIdx.x;
    if (row >= N_NODES) return;

    const float4* src = (const float4*)(unary + (size_t)row * N_UNARY);
    float4 x0 = src[0], x1 = src[1], x2 = src[2], x3 = src[3];
    float w0 = wu[0], w1 = wu[1], w2 = wu[2], w3 = wu[3];
    float d0, d1, d2;

    // clause 0: preds 0,1,2 signs (-,+,+)
    softmax3(-x0.x, x0.y, x0.z, w0, d0, d1, d2);
    x0.x -= d0; x0.y += d1; x0.z += d2;
    // clause 1: preds 3,4,5
    softmax3(-x0.w, x1.x, x1.y, w1, d0, d1, d2);
    x0.w -= d0; x1.x += d1; x1.y += d2;
    // clause 2: preds 6,7,8
    softmax3(-x1.z, x1.w, x2.x, w2, d0, d1, d2);
    x1.z -= d0; x1.w += d1; x2.x += d2;
    // clause 3: preds 9,10,11
    softmax3(-x2.y, x2.z, x2.w, w3, d0, d1, d2);
    x2.y -= d0; x2.z += d1; x2.w += d2;
    // preds 12..15 unchanged (x3)

    float4* o = (float4*)(out_u + (size_t)row * N_UNARY);
    o[0] = x0; o[1] = x1; o[2] = x2; o[3] = x3;
    float4* s = (float4*)(ws_u + (size_t)row * N_UNARY);
    s[0] = x0; s[1] = x1; s[2] = x2; s[3] = x3;
}

// Stage 2: per-edge clauses; binary tile staged into LDS via the Tensor Data Mover.
__global__ void kenn_edge_kernel(const float* __restrict__ binary,
                                 const float* __restrict__ wb,
                                 const int* __restrict__ edge_index,
                                 const float* __restrict__ ws_u,
                                 float* __restrict__ out_u,
                                 float* __restrict__ out_b) {
    __shared__ float4 tile[TILE_EDGES];   // 8 KB

    const int base = blockIdx.x * TILE_EDGES;

    // Wave 0 issues one TDM load of the block's contiguous binary tile into LDS.
    if (threadIdx.x < 32) {
        uint64_t gaddr = (uint64_t)(uintptr_t)(binary + (size_t)base * 4);
        uint32_t laddr = (uint32_t)(uintptr_t)(&tile[0]);   // LDS aperture: low 32 bits = LDS byte addr

        u32x4 g0;
        g0[0] = 1u;                                          // count=1 (valid), user mode
        g0[1] = laddr;                                       // lds_addr
        g0[2] = (uint32_t)gaddr;                             // global_addr[31:0]
        g0[3] = (uint32_t)((gaddr >> 32) & 0x01FFFFFFu)      // global_addr[56:32]
              | (2u << 30);                                  // type=2 ("image")

        i32x8 g1;
        g1[0] = 0x00020000;                                  // data_size=2 -> 4 bytes; mask=0
        g1[1] = (int)((TILE_ELEMS & 0xFFFF) << 16);          // tensor_dim0 lo16 (abar_addr=0)
        g1[2] = (int)(((TILE_ELEMS >> 16) & 0xFFFF)          // tensor_dim0 hi16
              | (1u << 16));                                 // tensor_dim1 = 1 (lo16)
        g1[3] = (int)((unsigned)TILE_ELEMS << 16);           // tile_dim0 = 2048 (dim1 hi16 = 0)
        g1[4] = 1;                                           // tile_dim1 = 1, tile_dim2 = 0
        g1[5] = TILE_ELEMS;                                  // tensor_dim0_stride lo32
        g1[6] = 0;                                           // stride0 hi16 | stride1 lo16
        g1[7] = 0;                                           // stride1 hi32

        i32x4 g2; g2[0] = 1; g2[1] = 0; g2[2] = 0; g2[3] = 0;  // tensor_dim2=1
        i32x4 g3; g3[0] = 0; g3[1] = 0; g3[2] = 0; g3[3] = 0;

        i32x8 g4;                                            // extra group (6-arg form); unused for 1-D tile
        g4[0] = 0; g4[1] = 0; g4[2] = 0; g4[3] = 0;
        g4[4] = 0; g4[5] = 0; g4[6] = 0; g4[7] = 0;

        __builtin_amdgcn_tensor_load_to_lds(g0, g1, g2, g3, g4, 0);
        __builtin_amdgcn_s_wait_tensorcnt(0);
    }
    __syncthreads();

    const float w0 = wb[0], w1 = wb[1], w2 = wb[2], w3 = wb[3];

#pragma unroll
    for (int k = 0; k < EPT; ++k) {
        const int el = threadIdx.x + k * EBLOCK;
        const int e  = base + el;

        const int i1 = edge_index[e];
        const int i2 = edge_index[N_EDGES + e];

        // L2-resident gathers of the first 4 predicates of each endpoint
        float4 a  = *(const float4*)(ws_u + (size_t)i1 * N_UNARY);
        float4 b  = *(const float4*)(ws_u + (size_t)i2 * N_UNARY);
        float4 bn = tile[el];

        float d0, d1, d2;
        float4 acc1, acc2;

        softmax3(-a.x, b.x, bn.x, w0, d0, d1, d2);
        acc1.x = -d0; acc2.x = d1; bn.x += d2;
        softmax3(-a.y, b.y, bn.y, w1, d0, d1, d2);
        acc1.y = -d0; acc2.y = d1; bn.y += d2;
        softmax3(-a.z, b.z, bn.z, w2, d0, d1, d2);
        acc1.z = -d0; acc2.z = d1; bn.z += d2;
        softmax3(-a.w, b.w, bn.w, w3, d0, d1, d2);
        acc1.w = -d0; acc2.w = d1; bn.w += d2;

        *(float4*)(out_b + (size_t)e * 4) = bn;

        float* p1 = out_u + (size_t)i1 * N_UNARY;
        float* p2 = out_u + (size_t)i2 * N_UNARY;
        atomicAdd(p1 + 0, acc1.x); atomicAdd(p1 + 1, acc1.y);
        atomicAdd(p1 + 2, acc1.z); atomicAdd(p1 + 3, acc1.w);
        atomicAdd(p2 + 0, acc2.x); atomicAdd(p2 + 1, acc2.y);
        atomicAdd(p2 + 2, acc2.z); atomicAdd(p2 + 3, acc2.w);
    }
}

extern "C" void kernel_launch(void* const* d_in, const int* in_sizes, int n_in,
                              void* d_out, int out_size, void* d_ws, size_t ws_size,
                              hipStream_t stream) {
    (void)in_sizes; (void)n_in; (void)out_size; (void)ws_size;

    const float* unary      = (const float*)d_in[0];   // [50000,16]
    const float* binary     = (const float*)d_in[1];   // [1600000,4]
    const float* wu         = (const float*)d_in[2];   // [4]
    const float* wb         = (const float*)d_in[3];   // [4]
    const int*   edge_index = (const int*)d_in[4];     // [2,1600000]

    float* out_u = (float*)d_out;                      // [50000,16]
    float* out_b = (float*)d_out + (size_t)N_NODES * N_UNARY; // [1600000,4]
    float* ws_u  = (float*)d_ws;                       // clean copy of u, 3.2 MB

    // Stage 1: nodes
    int nblocks = (N_NODES + 255) / 256;
    kenn_node_kernel<<<nblocks, 256, 0, stream>>>(unary, wu, out_u, ws_u);

    // Stage 2: edges (1.6M / 512 = 3125 blocks exactly)
    kenn_edge_kernel<<<N_EDGES / TILE_EDGES, EBLOCK, 0, stream>>>(
        binary, wb, edge_index, ws_u, out_u, out_b);
}